// GraphDecoder_14405320311212
// MI455X (gfx1250) — compile-verified
//
#include <hip/hip_runtime.h>

#define N_NODES 20000
#define LATD 64
#define HIDD 128
#define FEATD 64
#define NEDGE 640000
#define NLAYER 4

typedef __attribute__((ext_vector_type(2))) float v2f;
typedef __attribute__((ext_vector_type(8))) float v8f;

__device__ __forceinline__ void atomAddF(float* p, float v) {
    __hip_atomic_fetch_add(p, v, __ATOMIC_RELAXED, __HIP_MEMORY_SCOPE_AGENT);
}

// ---------------------------------------------------------------------------
// x = relu(z @ fc_w + fc_b), one thread per output element (coalesced per k)
// ---------------------------------------------------------------------------
__global__ __launch_bounds__(256) void fc_kernel(const float* __restrict__ z,
                                                 const float* __restrict__ fc_w,
                                                 const float* __restrict__ fc_b,
                                                 float* __restrict__ x) {
    __shared__ float zs[LATD];
    if (threadIdx.x < LATD) zs[threadIdx.x] = z[threadIdx.x];
    __syncthreads();
    const size_t NLAT = (size_t)N_NODES * LATD;
    size_t i = (size_t)blockIdx.x * blockDim.x + threadIdx.x;
    float s = fc_b[i];
#pragma unroll 8
    for (int k = 0; k < LATD; ++k)
        s += zs[k] * fc_w[(size_t)k * NLAT + i];
    x[i] = fmaxf(s, 0.0f);
}

// ---------------------------------------------------------------------------
// degree / normalization
// ---------------------------------------------------------------------------
__global__ void deg_init(float* __restrict__ deg) {
    int i = blockIdx.x * blockDim.x + threadIdx.x;
    if (i < N_NODES) deg[i] = 1.0f;   // self-loop contribution
}

__global__ void deg_count(const long long* __restrict__ col, float* __restrict__ deg) {
    int e = blockIdx.x * blockDim.x + threadIdx.x;
    if (e < NEDGE) atomAddF(&deg[(int)col[e]], 1.0f);
}

__global__ void dinv_kernel(const float* __restrict__ deg, float* __restrict__ dinv) {
    int i = blockIdx.x * blockDim.x + threadIdx.x;
    if (i < N_NODES) dinv[i] = rsqrtf(deg[i]);   // deg >= 1 always (self-loop)
}

// ---------------------------------------------------------------------------
// aggregation: out[i] = dinv[i]^2 * h[i] (+bias)  then  += norm_e * h[row_e]
// ---------------------------------------------------------------------------
__global__ __launch_bounds__(256) void agg_init(const float* __restrict__ h,
                                                const float* __restrict__ dinv,
                                                const float* __restrict__ bias,
                                                float* __restrict__ out, int cshift) {
    size_t t = (size_t)blockIdx.x * blockDim.x + threadIdx.x;
    int node = (int)(t >> cshift);
    int c    = (int)(t & ((1u << cshift) - 1u));
    float d = dinv[node];
    float v = d * d * h[t];
    if (bias) v += bias[c];
    out[t] = v;
}

__global__ __launch_bounds__(256) void agg_edges(const float* __restrict__ h,
                                                 const long long* __restrict__ row,
                                                 const long long* __restrict__ col,
                                                 const float* __restrict__ dinv,
                                                 float* __restrict__ out, int cshift) {
    int peShift = cshift - 2;                        // float4 chunks per edge
    int t = blockIdx.x * blockDim.x + threadIdx.x;
    int e  = t >> peShift;                           // uniform per group of C/4 lanes (perEdge | 32)
    int c4 = (t & ((1 << peShift) - 1)) << 2;
    if (e >= NEDGE) return;
    int r = (int)row[e];
    int c = (int)col[e];
    float nrm = dinv[r] * dinv[c];
    int C = 1 << cshift;
    const float4 v = *(const float4*)(h + (size_t)r * C + c4);
    float* o = out + (size_t)c * C + c4;
    atomAddF(o + 0, nrm * v.x);
    atomAddF(o + 1, nrm * v.y);
    atomAddF(o + 2, nrm * v.z);
    atomAddF(o + 3, nrm * v.w);
}

// ---------------------------------------------------------------------------
// GEMM via V_WMMA_F32_16X16X4_F32.
// One wave owns one 16-wide n-tile and sweeps MT consecutive 16-row m-tiles.
// B fragments (the whole 16-col weight strip) stay register-resident across
// the sweep; A loads are b64 at compile-time immediate offsets.
//   A[m][k]: lane = m&15 (+16 for K upper half), regs hold K = 2*(lane>>4)+{0,1}
//   B[k][n]: lane = n&15 (+16 for K upper half), regs hold same K pair
//   D[m][n]: vgpr v -> M = v + 8*(lane>>4), N = lane&15
// ---------------------------------------------------------------------------
template <int CIN, int COUT, int MT>
__global__ __launch_bounds__(128) void gemm_wmma(const float* __restrict__ A,
                                                 const float* __restrict__ W,
                                                 const float* __restrict__ bias,
                                                 float* __restrict__ out,
                                                 int doRelu) {
    constexpr int KF = CIN / 4;              // WMMA k-steps
    constexpr unsigned NT = COUT / 16;       // n-tiles
    unsigned lane = threadIdx.x & 31u;
    unsigned wave = threadIdx.x >> 5;
    unsigned task = blockIdx.x * 4u + wave;
    unsigned mg = task / NT;                 // m-group (MT tiles)
    unsigned nt = task % NT;
    unsigned m0 = mg * (16u * MT);
    unsigned l15 = lane & 15u;
    unsigned kh  = (lane >> 4) << 1;         // 0 or 2
    unsigned nB  = nt * 16u + l15;

    // Preload the 16-column B strip into registers (reused MT times).
    v2f bf[KF];
    const float* Wp = W + nB + kh * COUT;
#pragma unroll
    for (int k = 0; k < KF; ++k) {
        bf[k].x = Wp[(4u * k) * COUT];
        bf[k].y = Wp[(4u * k + 1u) * COUT];
    }
    float bv = bias ? bias[nB] : 0.0f;

    const float* Ap = A + (size_t)(m0 + l15) * CIN + kh;
    float* Op = out + (size_t)(m0 + ((lane >> 4) << 3)) * COUT + nB;

#pragma unroll 1
    for (int t = 0; t < MT; ++t) {
        v8f acc = {0.f, 0.f, 0.f, 0.f, 0.f, 0.f, 0.f, 0.f};
#pragma unroll
        for (int k = 0; k < KF; ++k) {
            v2f a = *(const v2f*)(Ap + 4u * k);     // b64, immediate offset
            acc = __builtin_amdgcn_wmma_f32_16x16x4_f32(
                false, a, false, bf[k], (short)0, acc, false, false);
        }
#pragma unroll
        for (int v = 0; v < 8; ++v) {
            float d = acc[v] + bv;
            if (doRelu) d = fmaxf(d, 0.0f);
            Op[(size_t)v * COUT] = d;               // b32, immediate offset
        }
        Ap += 16u * CIN;
        Op += 16u * COUT;
    }
}

// ---------------------------------------------------------------------------
extern "C" void kernel_launch(void* const* d_in, const int* in_sizes, int n_in,
                              void* d_out, int out_size, void* d_ws, size_t ws_size,
                              hipStream_t stream) {
    const float* z    = (const float*)d_in[0];
    const float* fc_w = (const float*)d_in[1];
    const float* fc_b = (const float*)d_in[2];
    const float* W0   = (const float*)d_in[3];
    const float* b0   = (const float*)d_in[4];
    const float* Wh   = (const float*)d_in[5];
    const float* bh   = (const float*)d_in[6];
    const float* Wl   = (const float*)d_in[7];
    const float* bl   = (const float*)d_in[8];
    const long long* ei   = (const long long*)d_in[9];
    const long long* rowp = ei;            // edge_index[0]
    const long long* colp = ei + NEDGE;    // edge_index[1]

    float* ws   = (float*)d_ws;
    float* deg  = ws;
    float* dinv = ws + N_NODES;
    float* bufX = ws + 2 * (size_t)N_NODES;                 // [N,64]
    float* bufA = bufX + (size_t)N_NODES * LATD;            // [N,128]
    float* bufB = bufA + (size_t)N_NODES * HIDD;            // [N,128]
    float* outp = (float*)d_out;                            // [N,64]

    // fc: z -> relu -> x0 [N,64]
    fc_kernel<<<(N_NODES * LATD) / 256, 256, 0, stream>>>(z, fc_w, fc_b, bufX);

    // gcn normalization
    deg_init<<<(N_NODES + 255) / 256, 256, 0, stream>>>(deg);
    deg_count<<<NEDGE / 256, 256, 0, stream>>>(colp, deg);
    dinv_kernel<<<(N_NODES + 255) / 256, 256, 0, stream>>>(deg, dinv);

    // m-groups: 1250 m-tiles = 250 groups of 5
    constexpr int MT = 5;
    constexpr int MG = (N_NODES / 16) / MT;                 // 250

    // layer 0 (64 -> 128): aggregate narrow side first (agg commutes with @W)
    agg_init<<<(N_NODES * LATD) / 256, 256, 0, stream>>>(bufX, dinv, nullptr, bufA, 6);
    agg_edges<<<(NEDGE * (LATD / 4)) / 256, 256, 0, stream>>>(bufX, rowp, colp, dinv, bufA, 6);
    gemm_wmma<LATD, HIDD, MT><<<MG * (HIDD / 16) / 4, 128, 0, stream>>>(bufA, W0, b0, bufB, 1);

    // hidden layers (128 -> 128)
    for (int l = 0; l < NLAYER; ++l) {
        agg_init<<<(N_NODES * HIDD) / 256, 256, 0, stream>>>(bufB, dinv, nullptr, bufA, 7);
        agg_edges<<<(NEDGE * (HIDD / 4)) / 256, 256, 0, stream>>>(bufB, rowp, colp, dinv, bufA, 7);
        gemm_wmma<HIDD, HIDD, MT><<<MG * (HIDD / 16) / 4, 128, 0, stream>>>(
            bufA, Wh + (size_t)l * HIDD * HIDD, bh + (size_t)l * HIDD, bufB, 1);
    }

    // final layer (128 -> 64): GEMM first (narrower output), then aggregate + bias
    gemm_wmma<HIDD, FEATD, MT><<<MG * (FEATD / 16) / 4, 128, 0, stream>>>(
        bufB, Wl, nullptr, bufA, 0);
    agg_init<<<(N_NODES * FEATD) / 256, 256, 0, stream>>>(bufA, dinv, bl, outp, 6);
    agg_edges<<<(NEDGE * (FEATD / 4)) / 256, 256, 0, stream>>>(bufA, rowp, colp, dinv, outp, 6);
}